// MultiHeadAttention_86337432584215
// MI455X (gfx1250) — compile-verified
//
#include <hip/hip_runtime.h>
#include <math.h>
#include <stdint.h>

// Problem constants (from the reference)
constexpr int kB   = 4;
constexpr int kN   = 512;
constexpr int kHid = 256;
constexpr int kH   = 8;
constexpr int kDk  = 32;
constexpr int kE   = 32;

typedef __attribute__((ext_vector_type(2))) float v2f;
typedef __attribute__((ext_vector_type(8))) float v8f;

#if __has_builtin(__builtin_amdgcn_tensor_load_to_lds)
#define HAVE_TDM 1
typedef __attribute__((ext_vector_type(4))) unsigned int v4u;
typedef __attribute__((ext_vector_type(4))) int v4i;
typedef __attribute__((ext_vector_type(8))) int v8i;
#else
#define HAVE_TDM 0
#endif

// CDNA5 V_WMMA_F32_16X16X4_F32 : D(16x16,f32) = A(16x4,f32) x B(4x16,f32) + C
// A layout (ISA 7.12.2): lanes 0-15 hold M=0..15 K={k0,k0+1}; lanes 16-31 hold
// M=0..15 K={k0+2,k0+3}.  B mirrors with N across lanes.  C/D: VGPR r holds
// row M=r (lanes 0-15) and M=r+8 (lanes 16-31), N = lane&15.
__device__ __forceinline__ v8f wmma_f32_k4(v2f a, v2f b, v8f c) {
  return __builtin_amdgcn_wmma_f32_16x16x4_f32(false, a, false, b, (short)0, c,
                                               false, false);
}

// ---------------------------------------------------------------------------
// Generic  out[M,Nn] = A[M,K] @ W[Nn,K]^T + bias   (row-major, f32 WMMA)
// One wave computes a 16x64 strip: 4 accumulators share one A fragment per
// K-step (4x less A traffic, 4 wmma per A load).  Requires Nn % 64 == 0.
// ---------------------------------------------------------------------------
__global__ __launch_bounds__(256) void gemm_nt_bias(
    const float* __restrict__ A, const float* __restrict__ W,
    const float* __restrict__ bias, float* __restrict__ out,
    int M, int K, int Nn) {
  const int lane = threadIdx.x & 31;
  const int wave = threadIdx.x >> 5;
  const int ntn  = Nn >> 6;                     // 64-wide N tiles
  const int tile = blockIdx.x * (blockDim.x >> 5) + wave;
  if (tile >= (M >> 4) * ntn) return;           // wave-uniform; EXEC stays full
  const int m0 = (tile / ntn) << 4;
  const int n0 = (tile % ntn) << 6;
  const int ml   = lane & 15;
  const int koff = (lane >> 4) << 1;
  const float* ar  = A + (m0 + ml) * K;
  const float* br0 = W + (n0 + ml) * K;         // B[k][n] = W[n][k]
  const float* br1 = br0 + 16 * K;
  const float* br2 = br0 + 32 * K;
  const float* br3 = br0 + 48 * K;
  v8f acc0 = {}, acc1 = {}, acc2 = {}, acc3 = {};
  for (int k0 = 0; k0 < K; k0 += 4) {
    v2f av, b0, b1, b2, b3;
    av.x = ar[k0 + koff];  av.y = ar[k0 + koff + 1];
    b0.x = br0[k0 + koff]; b0.y = br0[k0 + koff + 1];
    b1.x = br1[k0 + koff]; b1.y = br1[k0 + koff + 1];
    b2.x = br2[k0 + koff]; b2.y = br2[k0 + koff + 1];
    b3.x = br3[k0 + koff]; b3.y = br3[k0 + koff + 1];
    acc0 = wmma_f32_k4(av, b0, acc0);
    acc1 = wmma_f32_k4(av, b1, acc1);
    acc2 = wmma_f32_k4(av, b2, acc2);
    acc3 = wmma_f32_k4(av, b3, acc3);
  }
  const int rb = m0 + ((lane >> 4) << 3);
  const float bs0 = bias[n0 + ml];
  const float bs1 = bias[n0 + 16 + ml];
  const float bs2 = bias[n0 + 32 + ml];
  const float bs3 = bias[n0 + 48 + ml];
#pragma unroll
  for (int r = 0; r < 8; ++r) {
    float* orow = out + (rb + r) * Nn + n0 + ml;
    orow[0]  = acc0[r] + bs0;
    orow[16] = acc1[r] + bs1;
    orow[32] = acc2[r] + bs2;
    orow[48] = acc3[r] + bs3;
  }
}

// ---------------------------------------------------------------------------
// x[b,h,i,j] = mask( SCALE * qh[b,i,h,:]·kh[b,j,h,:] )
// ---------------------------------------------------------------------------
__global__ __launch_bounds__(32) void qk_mask(
    const float* __restrict__ qh, const float* __restrict__ kh,
    const float* __restrict__ adj, const int* __restrict__ use_adj_p,
    float* __restrict__ x) {
  const float SCALE = 0.17677669529663687f;     // 32^-0.5
  const float NEGV  = -9.0e15f;
  const int lane = threadIdx.x;
  const int i0 = blockIdx.x << 4, j0 = blockIdx.y << 4;
  const int bh = blockIdx.z;
  const int b = bh >> 3, h = bh & 7;
  const int ml = lane & 15, koff = (lane >> 4) << 1;
  const float* ar = qh + (b * kN + i0 + ml) * kHid + h * kDk;
  const float* br = kh + (b * kN + j0 + ml) * kHid + h * kDk;
  v8f acc = {};
#pragma unroll
  for (int k0 = 0; k0 < kDk; k0 += 4) {
    v2f av, bv;
    av.x = ar[k0 + koff]; av.y = ar[k0 + koff + 1];
    bv.x = br[k0 + koff]; bv.y = br[k0 + koff + 1];
    acc = wmma_f32_k4(av, bv, acc);
  }
  const int ua  = use_adj_p[0];
  const int col = j0 + ml;
#pragma unroll
  for (int r = 0; r < 8; ++r) {
    const int row = i0 + ((lane >> 4) << 3) + r;
    float v = acc[r] * SCALE;
    if (ua) {
      const float a = adj[(b * kN + row) * kN + col];
      v = (a > 0.f ? v : NEGV) * a;             // where(adj>0,x,NEG)*adj
    }
    x[((long)bh * kN + row) * kN + col] = v;
  }
}

// ---------------------------------------------------------------------------
// Edge branch, one wave handles 16 consecutive rows (b,i, j0..j0+15):
//   TDM copies the contiguous 16x32f edge_fea tile into this wave's LDS slab,
//   projT[b,h,j,i] = ef[b,i,j,:]·Wep[h,:] + bep[h]      (transposed store!)
//   t = ef + (x[b,:,j,i]·Wap^T + bap) * adj1[b,j,i]     -> LDS (in place)
//   edge_out[b,i,j,:] = t @ Woe^T                        (WMMA from LDS)
// ---------------------------------------------------------------------------
__global__ __launch_bounds__(256) void edge_kernel(
    const float* __restrict__ edge_fea, const float* __restrict__ x,
    const float* __restrict__ adj1, const float* __restrict__ Wap,
    const float* __restrict__ bap, const float* __restrict__ Wep,
    const float* __restrict__ bep, const float* __restrict__ Woe,
    float* __restrict__ projT, float* __restrict__ edge_out) {
  __shared__ float tls[8 * 16 * kE];            // 16 KB, per-wave 16x32 tile
  const int lane = threadIdx.x & 31;
  const int wave = threadIdx.x >> 5;
  const int w  = blockIdx.x * 8 + wave;         // chunk id
  const int j0 = (w & 31) << 4;
  const int i  = (w >> 5) & (kN - 1);
  const int b  = w >> 14;
  const int r  = lane & 15;                     // local row (j - j0)
  const int eh = lane >> 4;                     // which 16-wide e half
  const int j  = j0 + r;

  float vals[16];
#if HAVE_TDM
  {
    // D# (ISA 8.3/8.4): 2D tile, data_size=4B, one row of 512 elements (2 KB
    // contiguous) from edge_fea[b,i,j0,0] into this wave's LDS slab.
    const unsigned long long ga = (unsigned long long)(uintptr_t)(
        edge_fea + (((long)(b * kN + i)) * kN + j0) * kE);
    const unsigned lds_off = (unsigned)(wave * 16 * kE * 4);
    v4u g0;
    g0[0] = 1u;                                       // count=1, user D#
    g0[1] = lds_off;                                  // lds_addr
    g0[2] = (unsigned)(ga & 0xffffffffu);             // global_addr[31:0]
    g0[3] = (unsigned)((ga >> 32) & 0x01ffffffu) | 0x80000000u;  // addr | type=2
    v8i g1;
    g1[0] = 0x00020000;                 // workgroup_mask=0, data_size=4B
    g1[1] = (int)(512u << 16);          // tensor_dim0 = 512
    g1[2] = (int)(1u << 16);            // tensor_dim1 = 1
    g1[3] = (int)(512u << 16);          // tile_dim0 = 512
    g1[4] = 1;                          // tile_dim1 = 1
    g1[5] = 512;                        // tensor_dim0_stride = 512
    g1[6] = 0; g1[7] = 0;
    v4i z4 = {0, 0, 0, 0};
#if __clang_major__ >= 23
    v8i z8 = {0, 0, 0, 0, 0, 0, 0, 0};
    __builtin_amdgcn_tensor_load_to_lds(g0, g1, z4, z4, z8, 0);
#else
    __builtin_amdgcn_tensor_load_to_lds(g0, g1, z4, z4, 0);
#endif
    __builtin_amdgcn_s_wait_tensorcnt(0);       // wave-local fence on the DMA
  }
  {
    const float* lf = &tls[wave * 16 * kE + r * kE + eh * 16];
#pragma unroll
    for (int c = 0; c < 4; ++c) {
      const float4 t4 = ((const float4*)lf)[c];
      vals[4 * c + 0] = t4.x; vals[4 * c + 1] = t4.y;
      vals[4 * c + 2] = t4.z; vals[4 * c + 3] = t4.w;
    }
  }
#else
  {
    const float* efp =
        edge_fea + (((long)(b * kN + i)) * kN + j) * kE + eh * 16;
#pragma unroll
    for (int c = 0; c < 4; ++c) {
      const float4 t4 = ((const float4*)efp)[c];
      vals[4 * c + 0] = t4.x; vals[4 * c + 1] = t4.y;
      vals[4 * c + 2] = t4.z; vals[4 * c + 3] = t4.w;
    }
  }
#endif

  float xv[kH];
#pragma unroll
  for (int h = 0; h < kH; ++h)
    xv[h] = x[(((long)(b * kH + h)) * kN + j) * kN + i];
  const float a1 = adj1[(b * kN + j) * kN + i];

  // edge-bias projection (E->H), halves combined across lane^16
#pragma unroll
  for (int h = 0; h < kH; ++h) {
    float ph = 0.f;
#pragma unroll
    for (int el = 0; el < 16; ++el) ph += vals[el] * Wep[h * kE + eh * 16 + el];
    ph += __shfl_xor(ph, 16, 32);               // wave32 cross-half reduce
    if (eh == 0)
      projT[(((long)(b * kH + h)) * kN + j) * kN + i] = ph + bep[h];
  }

  // t = ef + xp*adj1  -> LDS (overwrites this wave's slab; wave-local order)
  float* tw = &tls[wave * 16 * kE];
#pragma unroll
  for (int el = 0; el < 16; ++el) {
    const int e = eh * 16 + el;
    float xp = bap[e];
#pragma unroll
    for (int h = 0; h < kH; ++h) xp += xv[h] * Wap[e * kH + h];
    tw[r * kE + e] = vals[el] + xp * a1;
  }
  __syncthreads();                              // DS writes visible to WMMA reads

  // t @ Woe^T : K=32, two N-tiles (e' 0..15 and 16..31), A shared
  const int ml = lane & 15, koff = (lane >> 4) << 1;
  const float* ar  = &tw[ml * kE];
  const float* br0 = Woe + ml * kE;
  const float* br1 = Woe + (16 + ml) * kE;
  v8f acc0 = {}, acc1 = {};
#pragma unroll
  for (int k0 = 0; k0 < kE; k0 += 4) {
    v2f av, b0, b1;
    av.x = ar[k0 + koff];  av.y = ar[k0 + koff + 1];
    b0.x = br0[k0 + koff]; b0.y = br0[k0 + koff + 1];
    b1.x = br1[k0 + koff]; b1.y = br1[k0 + koff + 1];
    acc0 = wmma_f32_k4(av, b0, acc0);
    acc1 = wmma_f32_k4(av, b1, acc1);
  }
#pragma unroll
  for (int rr = 0; rr < 8; ++rr) {
    const int jr = j0 + ((lane >> 4) << 3) + rr;
    const long obase = (((long)(b * kN + i)) * kN + jr) * kE;
    edge_out[obase + ml]      = acc0[rr];
    edge_out[obase + 16 + ml] = acc1[rr];
  }
}

// ---------------------------------------------------------------------------
// Fused dual softmax over j for one (b,h,i) row; writes p_long+p_local back
// in place so a single AV GEMM handles both branches.
// ---------------------------------------------------------------------------
__global__ __launch_bounds__(256) void softmax_combine(
    const float* __restrict__ attn_bias, const float* __restrict__ projT,
    const float* __restrict__ adj1, float* __restrict__ x) {
  __shared__ float red[256];
  const int row = blockIdx.x;                   // (b*H+h)*N + i
  const int i = row & (kN - 1);
  const int b = row / (kH * kN);
  const long base  = (long)row * kN;
  const long abase = ((long)(b * kN + i)) * kN;
  const int t = threadIdx.x;

  float u0, u1, w0, w1;
  {
    float xv = x[base + t], ab = attn_bias[base + t];
    float pj = projT[base + t], ad = adj1[abase + t];
    u0 = xv * ab; w0 = xv * pj * ad;
    xv = x[base + t + 256]; ab = attn_bias[base + t + 256];
    pj = projT[base + t + 256]; ad = adj1[abase + t + 256];
    u1 = xv * ab; w1 = xv * pj * ad;
  }
  auto rmax = [&](float v) -> float {
    red[t] = v; __syncthreads();
    for (int s = 128; s > 0; s >>= 1) {
      if (t < s) red[t] = fmaxf(red[t], red[t + s]);
      __syncthreads();
    }
    const float rv = red[0]; __syncthreads(); return rv;
  };
  auto rsum = [&](float v) -> float {
    red[t] = v; __syncthreads();
    for (int s = 128; s > 0; s >>= 1) {
      if (t < s) red[t] = red[t] + red[t + s];
      __syncthreads();
    }
    const float rv = red[0]; __syncthreads(); return rv;
  };
  const float mu = rmax(fmaxf(u0, u1));
  const float su = rsum(expf(u0 - mu) + expf(u1 - mu));
  const float mw = rmax(fmaxf(w0, w1));
  const float sw = rsum(expf(w0 - mw) + expf(w1 - mw));
  const float isu = 1.f / su, isw = 1.f / sw;
  x[base + t]       = expf(u0 - mu) * isu + expf(w0 - mw) * isw;
  x[base + t + 256] = expf(u1 - mu) * isu + expf(w1 - mw) * isw;
}

// ---------------------------------------------------------------------------
// y[b,i,h*Dk+d] = sum_j p[b,h,i,j] * vh[b,j,h*Dk+d]   (K = 512, WMMA)
// One wave computes both 16-wide d-tiles, sharing the A fragment.
// ---------------------------------------------------------------------------
__global__ __launch_bounds__(32) void attn_v(
    const float* __restrict__ p, const float* __restrict__ vh,
    float* __restrict__ y) {
  const int lane = threadIdx.x;
  const int i0 = blockIdx.x << 4;
  const int bh = blockIdx.z;
  const int b = bh >> 3, h = bh & 7;
  const int ml = lane & 15, koff = (lane >> 4) << 1;
  const float* ar = p + ((long)bh * kN + i0 + ml) * kN;
  v8f acc0 = {}, acc1 = {};
  for (int k0 = 0; k0 < kN; k0 += 4) {
    const float* vr0 = vh + (b * kN + k0 + koff) * kHid + h * kDk + ml;
    const float* vr1 = vr0 + kHid;              // k0+koff+1
    v2f av, b0, b1;
    av.x = ar[k0 + koff]; av.y = ar[k0 + koff + 1];
    b0.x = vr0[0];  b0.y = vr1[0];
    b1.x = vr0[16]; b1.y = vr1[16];
    acc0 = wmma_f32_k4(av, b0, acc0);
    acc1 = wmma_f32_k4(av, b1, acc1);
  }
#pragma unroll
  for (int r = 0; r < 8; ++r) {
    const int row = i0 + ((lane >> 4) << 3) + r;
    float* orow = y + (b * kN + row) * kHid + h * kDk + ml;
    orow[0]  = acc0[r];
    orow[16] = acc1[r];
  }
}

// ---------------------------------------------------------------------------
extern "C" void kernel_launch(void* const* d_in, const int* in_sizes, int n_in,
                              void* d_out, int out_size, void* d_ws,
                              size_t ws_size, hipStream_t stream) {
  (void)in_sizes; (void)n_in; (void)out_size; (void)ws_size;
  const float* q    = (const float*)d_in[0];
  const float* k    = (const float*)d_in[1];
  const float* v    = (const float*)d_in[2];
  const float* adj  = (const float*)d_in[3];
  const float* adj1 = (const float*)d_in[4];
  const float* ef   = (const float*)d_in[5];
  const float* ab   = (const float*)d_in[6];
  const float* Wq   = (const float*)d_in[7];
  const float* bq   = (const float*)d_in[8];
  const float* Wk   = (const float*)d_in[9];
  const float* bk   = (const float*)d_in[10];
  const float* Wv   = (const float*)d_in[11];
  const float* bv   = (const float*)d_in[12];
  const float* Wap  = (const float*)d_in[13];
  const float* bap  = (const float*)d_in[14];
  const float* Wep  = (const float*)d_in[15];
  const float* bep  = (const float*)d_in[16];
  const float* Wo   = (const float*)d_in[17];
  const float* bo   = (const float*)d_in[18];
  const float* Woe  = (const float*)d_in[19];
  const int* use_adj = (const int*)d_in[20];

  float* out      = (float*)d_out;
  float* edge_out = out + (size_t)kB * kN * kHid;

  float* ws = (float*)d_ws;
  const size_t QH = (size_t)kB * kN * kHid;        // 524288 floats
  const size_t XS = (size_t)kB * kH * kN * kN;     // 8388608 floats
  float* qh = ws;
  float* kh = qh + QH;
  float* vh = kh + QH;
  float* xb = vh + QH;                             // x, later p_long+p_local
  float* pT = xb + XS;                             // transposed edge bias
  float* yb = pT + XS;                             // attention output

  const int M = kB * kN;                           // 2048
  const int gemmBlocks = (M / 16) * (kHid / 64) / 8;  // 64 blocks of 8 waves

  gemm_nt_bias<<<dim3(gemmBlocks), dim3(256), 0, stream>>>(q, Wq, bq, qh, M, kHid, kHid);
  gemm_nt_bias<<<dim3(gemmBlocks), dim3(256), 0, stream>>>(k, Wk, bk, kh, M, kHid, kHid);
  gemm_nt_bias<<<dim3(gemmBlocks), dim3(256), 0, stream>>>(v, Wv, bv, vh, M, kHid, kHid);

  qk_mask<<<dim3(kN / 16, kN / 16, kB * kH), dim3(32), 0, stream>>>(
      qh, kh, adj, use_adj, xb);

  edge_kernel<<<dim3(kB * kN * (kN / 16) / 8), dim3(256), 0, stream>>>(
      ef, xb, adj1, Wap, bap, Wep, bep, Woe, pT, edge_out);

  softmax_combine<<<dim3(kB * kH * kN), dim3(256), 0, stream>>>(ab, pT, adj1, xb);

  attn_v<<<dim3(kN / 16, 1, kB * kH), dim3(32), 0, stream>>>(xb, vh, yb);

  gemm_nt_bias<<<dim3(gemmBlocks), dim3(256), 0, stream>>>(yb, Wo, bo, out, M, kHid, kHid);
}